// MatRNN_11605001634058
// MI455X (gfx1250) — compile-verified
//
#include <hip/hip_runtime.h>
#include <math.h>
#include <float.h>

typedef __attribute__((ext_vector_type(2))) float v2f;
typedef __attribute__((ext_vector_type(8))) float v8f;

#define BB   128
#define SS   128
#define QQ   32
#define KK   100
#define EMB  300
#define FEAT 2048
#define EPSC 1e-5f
#define KC   64       // k-chunk for LDS-staged GEMM
#define LDP  68       // padded LDS row stride (floats): 16B-aligned rows, no bank conflicts

static __device__ __forceinline__ v8f wmma4(v2f a, v2f b, v8f c) {
  // D = A(16x4 f32) x B(4x16 f32) + C(16x16 f32)
  return __builtin_amdgcn_wmma_f32_16x16x4_f32(
      /*neg_a=*/false, a, /*neg_b=*/false, b,
      /*c_mod=*/(short)0, c, /*reuse_a=*/false, /*reuse_b=*/false);
}

static __device__ __forceinline__ float sigm(float x) {
  return 1.0f / (1.0f + expf(-x));
}

// ---------------------------------------------------------------------------
// LDS-staged fp32 WMMA GEMM (Ka % 64 == 0):
//   C[M,N] = A[M,Ka] @ W[N,Ka]^T + bias[n] (+ gtab[gidx[row]*N + n])
// Block = 128 threads (4 waves); block computes a 64x64 tile. The W tile
// (shared by all 4 waves) is staged into LDS with async copies
// (global_load_async_to_lds_b128, ASYNCcnt), double-buffered so the next
// chunk's DMA overlaps the current chunk's WMMA work.
// ---------------------------------------------------------------------------
__global__ __launch_bounds__(128)
void gemm_wmma_lds(const float* __restrict__ A, const float* __restrict__ W,
                   const float* __restrict__ bias,
                   const int* __restrict__ gidx, const float* __restrict__ gtab,
                   float* __restrict__ C, int M, int N, int Ka)
{
  __shared__ __align__(16) float sw[2][64 * LDP];
  int tid = threadIdx.x;
  int wave = tid >> 5, lane = tid & 31;
  int l15 = lane & 15, hi = lane >> 4;
  int m0 = blockIdx.y * 64 + wave * 16;
  int n0 = blockIdx.x * 64;
  if (m0 >= M) return;

  const float* Arow = A + (size_t)(m0 + l15) * Ka;

  // Stage one 64(rows) x 64(k) W tile into sw[p] via async copies.
  // 1024 ids of 4 floats: id -> row = id>>4, k_in = (id&15)*4.
  auto issue = [&](int p, int kc) {
#pragma unroll
    for (int j = 0; j < 8; j++) {
      int id  = j * 128 + tid;
      int row = id >> 4;
      int kin = (id & 15) * 4;
      int nr  = n0 + row; if (nr >= N) nr = N - 1;   // clamp; store masked later
      const float* gp = W + (size_t)nr * Ka + kc + kin;
      unsigned lp = (unsigned)(size_t)&sw[p][row * LDP + kin];  // LDS addr = low 32 bits
      asm volatile("global_load_async_to_lds_b128 %0, %1, off"
                   :: "v"(lp), "v"(gp) : "memory");
    }
  };

  v8f acc[4] = {};
  int nchunk = Ka / KC;
  issue(0, 0);
  for (int c = 0; c < nchunk; c++) {
    int p = c & 1;
    asm volatile("s_wait_asynccnt 0x0" ::: "memory");  // own async writes landed
    __syncthreads();                                   // everyone's landed / reads done
    if (c + 1 < nchunk) issue(p ^ 1, (c + 1) * KC);    // overlap DMA with compute
    const float* bufw = &sw[p][0];
    int kc = c * KC;
#pragma unroll
    for (int k = 0; k < KC; k += 4) {
      int kk = k + 2 * hi;
      v2f a; a[0] = Arow[kc + kk]; a[1] = Arow[kc + kk + 1];
#pragma unroll
      for (int tt = 0; tt < 4; tt++) {
        const float* bp = bufw + (tt * 16 + l15) * LDP + kk;
        v2f b; b[0] = bp[0]; b[1] = bp[1];
        acc[tt] = wmma4(a, b, acc[tt]);
      }
    }
  }

#pragma unroll
  for (int tt = 0; tt < 4; tt++) {
    int nc = n0 + tt * 16 + l15;
    if (nc < N) {
      float badd = bias ? bias[nc] : 0.0f;
#pragma unroll
      for (int r = 0; r < 8; r++) {
        int row = m0 + r + 8 * hi;
        float v = acc[tt][r] + badd;
        if (gidx) v += gtab[(size_t)gidx[row] * N + nc];
        C[(size_t)row * N + nc] = v;
      }
    }
  }
}

// ---------------------------------------------------------------------------
// Generic fp32 WMMA GEMM (register-direct):  C = A @ W^T (+bias) (+addmat)
// Used for the Ka=600/300 GEMMs (Ka % 4 == 0). Layout as gemm_wmma_lds.
// ---------------------------------------------------------------------------
__global__ __launch_bounds__(128)
void gemm_wmma(const float* __restrict__ A, const float* __restrict__ W,
               const float* __restrict__ bias, const float* __restrict__ addmat,
               float* __restrict__ C, int M, int N, int Ka)
{
  int wave = threadIdx.x >> 5;
  int lane = threadIdx.x & 31;
  int l15 = lane & 15, hi = lane >> 4;
  int m0 = blockIdx.y * 64 + wave * 16;
  int n0 = blockIdx.x * 64;
  if (m0 >= M) return;

  const float* Arow = A + (size_t)(m0 + l15) * Ka;
  const float* Wr[4];
#pragma unroll
  for (int tt = 0; tt < 4; tt++) {
    int nc = n0 + tt * 16 + l15;
    if (nc >= N) nc = N - 1;
    Wr[tt] = W + (size_t)nc * Ka;
  }

  v8f acc[4] = {};
  for (int k = 0; k < Ka; k += 4) {
    int kk = k + 2 * hi;
    v2f a; a[0] = Arow[kk]; a[1] = Arow[kk + 1];
#pragma unroll
    for (int tt = 0; tt < 4; tt++) {
      v2f b; b[0] = Wr[tt][kk]; b[1] = Wr[tt][kk + 1];
      acc[tt] = wmma4(a, b, acc[tt]);
    }
  }

#pragma unroll
  for (int tt = 0; tt < 4; tt++) {
    int nc = n0 + tt * 16 + l15;
    if (nc < N) {
      float badd = bias ? bias[nc] : 0.0f;
#pragma unroll
      for (int r = 0; r < 8; r++) {
        int row = m0 + r + 8 * hi;
        float v = acc[tt][r] + badd;
        if (addmat) v += addmat[(size_t)row * N + nc];
        C[(size_t)row * N + nc] = v;
      }
    }
  }
}

// ---------------------------------------------------------------------------
// Fused LSTM step (both directions).  One wave per 16(batch) x 16(hidden)
// tile; computes all four gates with WMMA over K = [ h(300) | x_t(300) ],
// then the cell update with length-masking.  h double-buffered across steps.
// ---------------------------------------------------------------------------
__global__ __launch_bounds__(32)
void lstm_step(const float* __restrict__ cap, const float* __restrict__ cap_rev,
               const float* __restrict__ Whh_f, const float* __restrict__ Wih_f,
               const float* __restrict__ b_f,
               const float* __restrict__ Whh_b, const float* __restrict__ Wih_b,
               const float* __restrict__ b_b,
               const int* __restrict__ length,
               const float* __restrict__ h_prev, float* __restrict__ h_next,
               float* __restrict__ c_st,
               float* __restrict__ rnncat, float* __restrict__ out_b, int t)
{
  int dir = blockIdx.z;
  int by = blockIdx.y, bx = blockIdx.x;
  int lane = threadIdx.x & 31;
  int l15 = lane & 15, hi = lane >> 4;

  const float* x    = dir ? cap_rev : cap;
  const float* Whh  = dir ? Whh_b : Whh_f;
  const float* Wih  = dir ? Wih_b : Wih_f;
  const float* bias = dir ? b_b : b_f;

  int arow = by * 16 + l15;                                   // batch row (A frag)
  const float* hrow = h_prev + ((size_t)dir * BB + arow) * EMB;
  const float* xrow = x + ((size_t)arow * SS + t) * EMB;

  int nc = bx * 16 + l15;
  int ncl = nc < EMB ? nc : EMB - 1;
  const float* wh[4]; const float* wi[4];
#pragma unroll
  for (int g = 0; g < 4; g++) {
    wh[g] = Whh + ((size_t)(g * EMB + ncl)) * EMB;
    wi[g] = Wih + ((size_t)(g * EMB + ncl)) * EMB;
  }

  v8f acc[4] = {};
  // recurrent part: K = 0..299 from h_prev x Whh^T
  for (int k = 0; k < EMB; k += 4) {
    int kk = k + 2 * hi;
    v2f a; a[0] = hrow[kk]; a[1] = hrow[kk + 1];
#pragma unroll
    for (int g = 0; g < 4; g++) {
      v2f b; b[0] = wh[g][kk]; b[1] = wh[g][kk + 1];
      acc[g] = wmma4(a, b, acc[g]);
    }
  }
  // input part: K = 0..299 from x_t x Wih^T
  for (int k = 0; k < EMB; k += 4) {
    int kk = k + 2 * hi;
    v2f a; a[0] = xrow[kk]; a[1] = xrow[kk + 1];
#pragma unroll
    for (int g = 0; g < 4; g++) {
      v2f b; b[0] = wi[g][kk]; b[1] = wi[g][kk + 1];
      acc[g] = wmma4(a, b, acc[g]);
    }
  }

  int col = bx * 16 + l15;
  if (col < EMB) {
#pragma unroll
    for (int r = 0; r < 8; r++) {
      int brow = by * 16 + r + 8 * hi;
      float gi = sigm (acc[0][r] + bias[col]);
      float gf = sigm (acc[1][r] + bias[EMB + col]);
      float gg = tanhf(acc[2][r] + bias[2 * EMB + col]);
      float go = sigm (acc[3][r] + bias[3 * EMB + col]);
      size_t hix = ((size_t)dir * BB + brow) * EMB + col;
      float cold = c_st[hix];
      float cnew = gf * cold + gi * gg;
      float hnew = go * tanhf(cnew);
      bool  m    = t < length[brow];
      float hold = h_prev[hix];
      h_next[hix] = m ? hnew : hold;
      c_st[hix]   = m ? cnew : cold;
      float y = m ? hnew : 0.0f;
      if (dir == 0) rnncat[((size_t)brow * SS + t) * (2 * EMB) + col] = y;
      else          out_b [((size_t)brow * SS + t) * EMB + col]       = y;
    }
  }
}

// ---------------------------------------------------------------------------
// Attention: per (b, 16-row s-tile) block. 7 waves each do a 16x16 WMMA tile
// of att = hidden . k_emb^T / sqrt(300) into LDS; then per-row reduction:
// max(softmax) over K  ==  1 / sum_k exp(att - amax).
// ---------------------------------------------------------------------------
__global__ __launch_bounds__(256)
void att_kernel(const float* __restrict__ hidden, const float* __restrict__ kemb,
                float* __restrict__ max_att)
{
  __shared__ float lds[16 * 112];
  int b  = blockIdx.x >> 3;
  int s0 = (blockIdx.x & 7) * 16;
  int tid = threadIdx.x;
  int wv = tid >> 5, lane = tid & 31, l15 = lane & 15, hi = lane >> 4;

  if (wv < 7) {
    const float* Arow = hidden + ((size_t)(b * SS + s0 + l15)) * EMB;
    int kc = wv * 16 + l15;
    int kcl = kc < KK ? kc : KK - 1;
    const float* Brow = kemb + ((size_t)(b * KK + kcl)) * EMB;
    v8f acc = {};
    for (int k = 0; k < EMB; k += 4) {
      int kk = k + 2 * hi;
      v2f a; a[0] = Arow[kk]; a[1] = Arow[kk + 1];
      v2f bb; bb[0] = Brow[kk]; bb[1] = Brow[kk + 1];
      acc = wmma4(a, bb, acc);
    }
    const float scale = 0.05773502691896258f;  // 1/sqrt(300)
#pragma unroll
    for (int r = 0; r < 8; r++)
      lds[(r + 8 * hi) * 112 + wv * 16 + l15] = acc[r] * scale;
  }
  __syncthreads();
  if (tid < 16) {
    float amax = -FLT_MAX;
    for (int k = 0; k < KK; k++) { float v = lds[tid * 112 + k]; amax = v > amax ? v : amax; }
    float Z = 0.0f;
    for (int k = 0; k < KK; k++) Z += expf(lds[tid * 112 + k] - amax);
    max_att[b * SS + s0 + tid] = 1.0f / Z;
  }
}

// ---------------------------------------------------------------------------
// Span-masked softmax over S + weighted pooling p = w @ hidden. One block per
// batch row; w[32][128] in LDS; each thread owns one embedding dim.
// ---------------------------------------------------------------------------
__global__ __launch_bounds__(320)
void wp_kernel(const float* __restrict__ max_att, const int* __restrict__ span,
               const float* __restrict__ hidden, float* __restrict__ p)
{
  __shared__ float ma[SS];
  __shared__ float wq[QQ * SS];
  int b = blockIdx.x, tid = threadIdx.x;
  if (tid < SS) ma[tid] = max_att[b * SS + tid];
  __syncthreads();
  if (tid < QQ) {
    const int* msk = span + ((size_t)b * QQ + tid) * SS;
    float amax = -FLT_MAX;
    for (int s = 0; s < SS; s++) if (msk[s]) { float v = ma[s]; amax = v > amax ? v : amax; }
    if (amax == -FLT_MAX) {                // no valid span: reference -> uniform
      float u = 1.0f / SS;
      for (int s = 0; s < SS; s++) wq[tid * SS + s] = u;
    } else {
      float Z = 0.0f;
      for (int s = 0; s < SS; s++) {
        float e = msk[s] ? expf(ma[s] - amax) : 0.0f;
        wq[tid * SS + s] = e; Z += e;
      }
      float iz = 1.0f / Z;
      for (int s = 0; s < SS; s++) wq[tid * SS + s] *= iz;
    }
  }
  __syncthreads();
  if (tid < EMB) {
    float acc[QQ];
#pragma unroll
    for (int q = 0; q < QQ; q++) acc[q] = 0.0f;
    for (int s = 0; s < SS; s++) {
      float hd = hidden[((size_t)b * SS + s) * EMB + tid];
#pragma unroll
      for (int q = 0; q < QQ; q++) acc[q] += wq[q * SS + s] * hd;
    }
    for (int q = 0; q < QQ; q++)
      p[((size_t)(b * QQ + q)) * EMB + tid] = acc[q];
  }
}

// ---------------------------------------------------------------------------
// Small helpers
// ---------------------------------------------------------------------------
__global__ void zero_kernel(float* p, int n) {
  int i = blockIdx.x * blockDim.x + threadIdx.x;
  if (i < n) p[i] = 0.0f;
}

__global__ void combine_kernel(const float* Wp, const float* Wm,
                               const float* bp, const float* bm,
                               float* Wc, float* bc) {
  int i = blockIdx.x * blockDim.x + threadIdx.x;
  if (i < EMB * EMB) Wc[i] = Wp[i] + EPSC * Wm[i];
  if (i < EMB)       bc[i] = bp[i] + EPSC * bm[i];
}

// cap[b,s,:] = emb[caption_id[b,s]];  cap_rev[b,s,:] = emb[caption_id[b,rev(b,s)]]
__global__ __launch_bounds__(320)
void prep_gather(const int* __restrict__ cid, const int* __restrict__ length,
                 const float* __restrict__ emb,
                 float* __restrict__ cap, float* __restrict__ cap_rev)
{
  int s = blockIdx.x, b = blockIdx.y, d = threadIdx.x;
  if (d >= EMB) return;
  int len = length[b];
  cap[((size_t)b * SS + s) * EMB + d] = emb[(size_t)cid[b * SS + s] * EMB + d];
  int rev = (s < len) ? (len - 1 - s) : s;
  cap_rev[((size_t)b * SS + s) * EMB + d] = emb[(size_t)cid[b * SS + rev] * EMB + d];
}

// rnncat[b,t,300+d] = out_b_scan[b, rev(b,t), d]
__global__ __launch_bounds__(320)
void gather_outb(const float* __restrict__ out_b, const int* __restrict__ length,
                 float* __restrict__ rnncat)
{
  int t = blockIdx.x, b = blockIdx.y, d = threadIdx.x;
  if (d >= EMB) return;
  int len = length[b];
  int rev = (t < len) ? (len - 1 - t) : t;
  rnncat[((size_t)b * SS + t) * (2 * EMB) + EMB + d] =
      out_b[((size_t)b * SS + rev) * EMB + d];
}

// ---------------------------------------------------------------------------
extern "C" void kernel_launch(void* const* d_in, const int* in_sizes, int n_in,
                              void* d_out, int out_size, void* d_ws, size_t ws_size,
                              hipStream_t stream) {
  (void)in_sizes; (void)n_in; (void)out_size; (void)ws_size;
  const int*   caption_id = (const int*)d_in[0];
  const int*   span       = (const int*)d_in[1];
  const int*   length     = (const int*)d_in[2];
  const int*   label      = (const int*)d_in[3];
  const float* feature    = (const float*)d_in[4];
  const float* emb        = (const float*)d_in[5];
  const float* Wf         = (const float*)d_in[6];
  const float* bf         = (const float*)d_in[7];
  const float* Wih_f      = (const float*)d_in[8];
  const float* Whh_f      = (const float*)d_in[9];
  const float* b_f        = (const float*)d_in[10];
  const float* Wih_b      = (const float*)d_in[11];
  const float* Whh_b      = (const float*)d_in[12];
  const float* b_b        = (const float*)d_in[13];
  const float* Wrnn       = (const float*)d_in[14];
  const float* brnn       = (const float*)d_in[15];
  const float* Wp         = (const float*)d_in[16];
  const float* bp         = (const float*)d_in[17];
  const float* Wm         = (const float*)d_in[18];
  const float* bm         = (const float*)d_in[19];

  float* p_emb = (float*)d_out;                          // [B,Q,EMB]
  float* k_emb = p_emb + (size_t)BB * QQ * EMB;          // [B,K,EMB]

  float* w = (float*)d_ws;
  float* cap     = w;  w += (size_t)BB * SS * EMB;       // hidden written in place
  float* cap_rev = w;  w += (size_t)BB * SS * EMB;
  float* rnncat  = w;  w += (size_t)BB * SS * 2 * EMB;
  float* out_b   = w;  w += (size_t)BB * SS * EMB;
  float* pbuf    = w;  w += (size_t)BB * QQ * EMB;
  float* h_buf   = w;  w += (size_t)2 * 2 * BB * EMB;    // [parity][dir][B][EMB]
  float* c_buf   = w;  w += (size_t)2 * BB * EMB;        // [dir][B][EMB]
  float* max_att = w;  w += (size_t)BB * SS;
  float* Wc      = w;  w += (size_t)EMB * EMB;
  float* bc      = w;  w += 320;

  // zero h (both parities) and c — contiguous region
  int nz = (4 + 2) * BB * EMB;
  zero_kernel<<<(nz + 255) / 256, 256, 0, stream>>>(h_buf, nz);

  combine_kernel<<<(EMB * EMB + 255) / 256, 256, 0, stream>>>(Wp, Wm, bp, bm, Wc, bc);

  prep_gather<<<dim3(SS, BB), 320, 0, stream>>>(caption_id, length, emb, cap, cap_rev);

  // k_emb = emb[label] + feature @ Wf^T + bf   (M=12800, N=300, K=2048)
  // LDS-staged with async copies (Ka % 64 == 0)
  gemm_wmma_lds<<<dim3(5, (BB * KK) / 64), 128, 0, stream>>>(
      feature, Wf, bf, label, emb, k_emb, BB * KK, EMB, FEAT);

  // bidirectional masked LSTM scan
  for (int t = 0; t < SS; t++) {
    const float* hprev = h_buf + (size_t)(t & 1) * 2 * BB * EMB;
    float*       hnext = h_buf + (size_t)((t + 1) & 1) * 2 * BB * EMB;
    lstm_step<<<dim3(19, BB / 16, 2), 32, 0, stream>>>(
        cap, cap_rev, Whh_f, Wih_f, b_f, Whh_b, Wih_b, b_b,
        length, hprev, hnext, c_buf, rnncat, out_b, t);
  }

  gather_outb<<<dim3(SS, BB), 320, 0, stream>>>(out_b, length, rnncat);

  // hidden = cap + rnncat @ Wrnn^T + brnn   (in-place into cap)
  gemm_wmma<<<dim3(5, (BB * SS) / 64), 128, 0, stream>>>(
      rnncat, Wrnn, brnn, cap, cap, BB * SS, EMB, 2 * EMB);

  att_kernel<<<BB * (SS / 16), 256, 0, stream>>>(cap, k_emb, max_att);

  wp_kernel<<<BB, 320, 0, stream>>>(max_att, span, cap, pbuf);

  // p_emb = p @ (Wp + eps*Wm)^T + (bp + eps*bm)
  gemm_wmma<<<dim3(5, (BB * QQ) / 64), 128, 0, stream>>>(
      pbuf, Wc, bc, nullptr, p_emb, BB * QQ, EMB, EMB);
}